// D2VEncoder_24446953849432
// MI455X (gfx1250) — compile-verified
//
#include <hip/hip_runtime.h>
#include <cstdint>

// Problem constants (match reference)
#define BB    16
#define TT    4096
#define DD    768
#define LQ    (DD / 4)      // 192 vec4 per row
#define ROWS  (BB * TT)     // 65536
#define SPAN  10
#define PMASK 0.065f
#define RPB   8             // rows per block in apply kernel

// Native 16-byte vector type (accepted by __builtin_nontemporal_*)
typedef float v4f __attribute__((ext_vector_type(4)));

// ---------------------------------------------------------------------------
// Kernel 1: span mask flags + the [B,T] -inf/0 mask output.
// One thread per (b,t). masked(t) = OR_{i=max(0,t-9)}^{t} (rand[b,i] < P).
// Fixed 10-wide clamped window: all 10 loads are independent (single wait),
// boundary duplicates are no-ops under OR.
// ---------------------------------------------------------------------------
__global__ __launch_bounds__(256) void span_flags_kernel(
    const float* __restrict__ rv,
    unsigned char* __restrict__ flags,
    float* __restrict__ mask_out)
{
    const int idx = blockIdx.x * 256 + threadIdx.x;   // 0 .. ROWS-1
    if (idx >= ROWS) return;
    const int t  = idx & (TT - 1);
    const int rb = idx - t;                           // row base b*TT

    float v[SPAN];
#pragma unroll
    for (int k = 0; k < SPAN; ++k) {
        int i = t - k;
        i = i < 0 ? 0 : i;                            // clamp (dup reads ok)
        v[k] = rv[rb + i];
    }
    unsigned m = 0u;
#pragma unroll
    for (int k = 0; k < SPAN; ++k)
        m |= (v[k] < PMASK) ? 1u : 0u;

    flags[idx]    = (unsigned char)m;
    mask_out[idx] = m ? -__builtin_inff() : 0.0f;
}

// ---------------------------------------------------------------------------
// Kernel 2: streaming apply. 192 threads = one vec4 lane per D-column chunk,
// 8 rows per block. mask_token (3 KB) is staged into LDS with the gfx1250
// async copy (global_load_async_to_lds_b128, tracked by ASYNCcnt). Each lane
// stages exactly the 16 B it later reads, so a per-wave s_wait_asynccnt 0 is
// sufficient synchronization (no cross-wave barrier needed).
// The 8 row flags come from one aligned 8-byte uniform load; the per-row
// branch is divergence-free, and masked rows never touch x in HBM.
// Streaming accesses are non-temporal (no reuse; keep L2 clean).
// ---------------------------------------------------------------------------
__global__ __launch_bounds__(LQ) void apply_mask_kernel(
    const v4f* __restrict__ x,
    const v4f* __restrict__ token,
    const unsigned long long* __restrict__ flags8,   // flags viewed as u64
    v4f* __restrict__ out)
{
    const int tid = threadIdx.x;                      // 0 .. 191
    __shared__ v4f tok[LQ];

    // --- CDNA5 async memory -> LDS copy of the mask token ---
    {
        unsigned           lds_off = (unsigned)(uintptr_t)(&tok[tid]);
        unsigned long long gaddr   = (unsigned long long)(uintptr_t)(token + tid);
        asm volatile("global_load_async_to_lds_b128 %0, %1, off"
                     :: "v"(lds_off), "v"(gaddr)
                     : "memory");
        asm volatile("s_wait_asynccnt 0" ::: "memory");
    }
    const v4f tv = tok[tid];                          // ds_load_b128

    const int row0 = blockIdx.x * RPB;
    const unsigned long long f8 = flags8[blockIdx.x]; // 8 flags, s_load_b64

#pragma unroll
    for (int r = 0; r < RPB; ++r) {
        const int ofs = (row0 + r) * LQ + tid;        // vec4 index, < 2^24
        if ((f8 >> (8 * r)) & 1ull) {                 // uniform SALU test
            __builtin_nontemporal_store(tv, &out[ofs]);        // skip x read
        } else {
            v4f xv = __builtin_nontemporal_load(&x[ofs]);      // b128 NT stream
            __builtin_nontemporal_store(xv, &out[ofs]);
        }
    }
}

// ---------------------------------------------------------------------------
// Host-side launch. Inputs: x [B,T,D] f32, mask_token [D] f32,
// rand_vals [B,T] f32. Output: x_out (B*T*D floats) ++ mask (B*T floats).
// Workspace: ROWS bytes of flags (d_ws is 256-B aligned => u64 view is safe).
// ---------------------------------------------------------------------------
extern "C" void kernel_launch(void* const* d_in, const int* in_sizes, int n_in,
                              void* d_out, int out_size, void* d_ws, size_t ws_size,
                              hipStream_t stream)
{
    const float* x     = (const float*)d_in[0];
    const float* token = (const float*)d_in[1];
    const float* rv    = (const float*)d_in[2];

    float* xout    = (float*)d_out;
    float* maskout = xout + (size_t)ROWS * DD;
    unsigned char* flags = (unsigned char*)d_ws;

    span_flags_kernel<<<ROWS / 256, 256, 0, stream>>>(rv, flags, maskout);
    apply_mask_kernel<<<ROWS / RPB, LQ, 0, stream>>>(
        (const v4f*)x, (const v4f*)token,
        (const unsigned long long*)flags, (v4f*)xout);
}